// E88FLAHybrid_30726196035860
// MI455X (gfx1250) — compile-verified
//
#include <hip/hip_runtime.h>

// ---- problem constants (from reference) ----
#define T_STEPS 2048
#define BB      4
#define DIMX    1024
#define NH      8
#define KEY_DIM 512
#define VAL_DIM 512
#define QKV_N   1536
#define ROWS    8192   // T*B

typedef __attribute__((ext_vector_type(2))) float v2f;
typedef __attribute__((ext_vector_type(4))) float v4f;
typedef __attribute__((ext_vector_type(8))) float v8f;

// ------------------------------------------------------------------
// FP32 GEMM via V_WMMA_F32_16X16X4_F32: C[M,N] = X[M,K] * W[N,K]^T
// One wave computes a 16x64 strip (4 N-tiles). K is processed 16 at a
// time: all 20 fragment loads (4 A + 16 B) are issued before the 16
// WMMAs so the scheduler can clause the loads and overlap them with
// the matrix pipe instead of draining loadcnt per WMMA.
// A layout (16x4 f32): lane = m + 16*(k/2), vgpr = k%2
// B layout (4x16 f32): lane = n + 16*(k/2), vgpr = k%2  (B = W^T)
// C/D layout: vgpr j -> M = tm + j + 8*(lane>=16), N = tn + (lane&15)
// ------------------------------------------------------------------
__global__ __launch_bounds__(256)
void gemm_f32_wmma(const float* __restrict__ X, const float* __restrict__ W,
                   float* __restrict__ C, int N, int K) {
  const int wave    = blockIdx.x * 8 + (threadIdx.x >> 5);
  const int lane    = threadIdx.x & 31;
  const int tilesN4 = N >> 6;                 // strips of 4 N-tiles
  const int tm = (wave / tilesN4) << 4;
  const int tn = (wave % tilesN4) << 6;
  const int r  = lane & 15;
  const int kh = (lane >> 4) << 1;            // 0 or 2
  const float* xr  = X + (size_t)(tm + r) * K + kh;
  const float* wr0 = W + (size_t)(tn + r) * K + kh;
  const float* wr1 = wr0 + (size_t)16 * K;
  const float* wr2 = wr0 + (size_t)32 * K;
  const float* wr3 = wr0 + (size_t)48 * K;

  v8f acc0, acc1, acc2, acc3;
#pragma unroll
  for (int j = 0; j < 8; ++j) { acc0[j] = 0.f; acc1[j] = 0.f; acc2[j] = 0.f; acc3[j] = 0.f; }

  for (int k0 = 0; k0 < K; k0 += 16) {
    v2f a[4], b0[4], b1[4], b2[4], b3[4];
#pragma unroll
    for (int j = 0; j < 4; ++j) a[j]  = *(const v2f*)(xr  + k0 + 4 * j);
#pragma unroll
    for (int j = 0; j < 4; ++j) b0[j] = *(const v2f*)(wr0 + k0 + 4 * j);
#pragma unroll
    for (int j = 0; j < 4; ++j) b1[j] = *(const v2f*)(wr1 + k0 + 4 * j);
#pragma unroll
    for (int j = 0; j < 4; ++j) b2[j] = *(const v2f*)(wr2 + k0 + 4 * j);
#pragma unroll
    for (int j = 0; j < 4; ++j) b3[j] = *(const v2f*)(wr3 + k0 + 4 * j);

#pragma unroll
    for (int j = 0; j < 4; ++j) {
      acc0 = __builtin_amdgcn_wmma_f32_16x16x4_f32(false, a[j], false, b0[j], (short)0, acc0, false, false);
      acc1 = __builtin_amdgcn_wmma_f32_16x16x4_f32(false, a[j], false, b1[j], (short)0, acc1, false, false);
      acc2 = __builtin_amdgcn_wmma_f32_16x16x4_f32(false, a[j], false, b2[j], (short)0, acc2, false, false);
      acc3 = __builtin_amdgcn_wmma_f32_16x16x4_f32(false, a[j], false, b3[j], (short)0, acc3, false, false);
    }
  }

  const int mb = tm + ((lane >> 4) << 3);
#pragma unroll
  for (int j = 0; j < 8; ++j) {
    C[(size_t)(mb + j) * N + tn + r]      = acc0[j];
    C[(size_t)(mb + j) * N + tn + 16 + r] = acc1[j];
    C[(size_t)(mb + j) * N + tn + 32 + r] = acc2[j];
    C[(size_t)(mb + j) * N + tn + 48 + r] = acc3[j];
  }
}

// ------------------------------------------------------------------
// L2-normalize one 64-float group (a q-head or k-head) per 32-lane block.
// grid = ROWS*16 : gg<8 -> q head gg, gg>=8 -> k head gg-8.
// ------------------------------------------------------------------
__global__ __launch_bounds__(32)
void l2norm_kernel(float* __restrict__ qkv) {
  const int g  = blockIdx.x;
  const int m  = g >> 4;
  const int gg = g & 15;
  const int off = (gg < 8) ? (gg << 6) : (KEY_DIM + ((gg - 8) << 6));
  float* p = qkv + (size_t)m * QKV_N + off;
  const int lane = threadIdx.x;
  v2f val = *(v2f*)(p + (lane << 1));
  float ss = val.x * val.x + val.y * val.y;
#pragma unroll
  for (int o = 16; o > 0; o >>= 1) ss += __shfl_xor(ss, o, 32);
  const float rs = rsqrtf(ss + 1e-6f);
  val.x *= rs; val.y *= rs;
  *(v2f*)(p + (lane << 1)) = val;
}

// ------------------------------------------------------------------
// SiLU over the v slice of qkv (in place).
// ------------------------------------------------------------------
__global__ __launch_bounds__(256)
void silu_kernel(float* __restrict__ qkv) {
  const int idx = blockIdx.x * 256 + threadIdx.x;   // 0 .. ROWS*512-1
  const int m = idx >> 9;
  const int c = idx & 511;
  float* p = qkv + (size_t)m * QKV_N + 1024 + c;
  const float x = *p;
  *p = x / (1.0f + __expf(-x));
}

// ------------------------------------------------------------------
// decay[m,h] = exp(-exp(A_log[h]) * softplus(x[m]·W_a[h] + dt_bias[h]))
// ------------------------------------------------------------------
__global__ __launch_bounds__(256)
void decay_kernel(const float* __restrict__ x, const float* __restrict__ Wa,
                  const float* __restrict__ A_log, const float* __restrict__ dt_bias,
                  float* __restrict__ decay) {
  const int m    = blockIdx.x;
  const int h    = threadIdx.x >> 5;
  const int lane = threadIdx.x & 31;
  const float* xr = x  + (size_t)m * DIMX;
  const float* wr = Wa + (size_t)h * DIMX;
  float s = 0.f;
  for (int d = lane; d < DIMX; d += 32) s = fmaf(xr[d], wr[d], s);
#pragma unroll
  for (int o = 16; o > 0; o >>= 1) s += __shfl_xor(s, o, 32);
  if (lane == 0) {
    const float a  = s + dt_bias[h];
    const float sp = (a > 20.f) ? a : log1pf(__expf(a));
    decay[m * NH + h] = __expf(-__expf(A_log[h]) * sp);
  }
}

__device__ __forceinline__ float fast_tanh(float x) {
#if __has_builtin(__builtin_amdgcn_tanhf)
  return __builtin_amdgcn_tanhf(x);            // v_tanh_f32 on gfx1250
#else
  const float e = __expf(2.0f * x);
  return 1.0f - 2.0f / (e + 1.0f);
#endif
}

// ------------------------------------------------------------------
// Sequential scan. One workgroup per (b,h); S[64,64] in registers
// (256 threads x 16 f32). Next timestep's q/k/v/decay is prefetched
// into a double-buffered LDS stage via GLOBAL_LOAD_ASYNC_TO_LDS
// (ASYNCcnt), overlapping global latency with the tanh/FMA chain.
// stage layout per buffer: [0..63]=q, [64..127]=k, [128..191]=v, [192]=decay
// ------------------------------------------------------------------
__global__ __launch_bounds__(256)
void scan_kernel(const float* __restrict__ qkv, const float* __restrict__ decay,
                 float* __restrict__ outs) {
  const int bh  = blockIdx.x;        // b*8 + h, 32 blocks
  const int b   = bh >> 3;
  const int h   = bh & 7;
  const int tid = threadIdx.x;
  const int hv  = tid & 63;
  const int n0  = (tid >> 6) << 4;   // 0,16,32,48

  float S[16];
#pragma unroll
  for (int i = 0; i < 16; ++i) S[i] = 0.f;

  __shared__ __align__(16) float stage[2][208];
  __shared__ float red[256];

  const int chunk = tid >> 4;                 // 0=q,1=k,2=v (for tid<48)
  const int ci    = (tid & 15) << 2;          // float index within chunk
  const int goff  = (chunk == 0) ? 0 : (chunk == 1 ? KEY_DIM : 1024);

  auto issue = [&](int buf, int t) {
    const size_t row = (size_t)(t * BB + b) * QKV_N + (size_t)h * 64;
    if (tid < 48) {
      const float* g = qkv + row + goff + ci;
      const unsigned l = (unsigned)(size_t)&stage[buf][chunk * 64 + ci];
      asm volatile("global_load_async_to_lds_b128 %0, %1, off"
                   :: "v"(l), "v"(g) : "memory");
    } else if (tid == 48) {
      const float* g = decay + (t * BB + b) * NH + h;
      const unsigned l = (unsigned)(size_t)&stage[buf][192];
      asm volatile("global_load_async_to_lds_b32 %0, %1, off"
                   :: "v"(l), "v"(g) : "memory");
    }
  };

  issue(0, 0);

  for (int t = 0; t < T_STEPS; ++t) {
    const int cur = t & 1;
    // waves 0-1 wait for their async stage; others have ASYNCcnt==0
    asm volatile("s_wait_asynccnt 0x0" ::: "memory");
    __syncthreads();                          // stage[cur] visible to all

    const float* st = stage[cur];
    v4f qq[4], kq[4];
#pragma unroll
    for (int j = 0; j < 4; ++j) {
      qq[j] = *(const v4f*)(st + n0 + 4 * j);
      kq[j] = *(const v4f*)(st + 64 + n0 + 4 * j);
    }
    const float vh = st[128 + hv];
    const float d  = st[192];

    // prefetch next step into the other buffer (no conflict with reads)
    if (t + 1 < T_STEPS) issue(cur ^ 1, t + 1);

    float partial = 0.f;
#pragma unroll
    for (int j = 0; j < 4; ++j)
#pragma unroll
      for (int e = 0; e < 4; ++e) {
        const int i = 4 * j + e;
        const float s = fast_tanh(fmaf(d, S[i], kq[j][e] * vh));
        S[i] = s;
        partial = fmaf(qq[j][e], s, partial);
      }

    red[tid] = partial;
    __syncthreads();                          // also fences stage[cur] WAR
    if (tid < 64)
      outs[(size_t)(t * 32 + bh) * 64 + tid] =
          red[tid] + red[tid + 64] + red[tid + 128] + red[tid + 192];
    __syncthreads();
  }
}

// ------------------------------------------------------------------
extern "C" void kernel_launch(void* const* d_in, const int* in_sizes, int n_in,
                              void* d_out, int out_size, void* d_ws, size_t ws_size,
                              hipStream_t stream) {
  const float* x       = (const float*)d_in[0];   // [2048,4,1024]
  const float* W_qkv   = (const float*)d_in[1];   // [1536,1024]
  const float* W_a     = (const float*)d_in[2];   // [8,1024]
  const float* A_log   = (const float*)d_in[3];   // [8]
  const float* dt_bias = (const float*)d_in[4];   // [8]
  const float* W_out   = (const float*)d_in[5];   // [1024,512]
  float* out = (float*)d_out;                     // [2048,4,1024]

  float* qkv  = (float*)d_ws;                          // ROWS*1536
  float* outs = qkv  + (size_t)ROWS * QKV_N;           // ROWS*512
  float* dec  = outs + (size_t)ROWS * VAL_DIM;         // ROWS*8

  // 1) qkv = x @ W_qkv^T : waves = (8192/16)*(1536/64) = 12288 -> 1536 blocks
  gemm_f32_wmma<<<1536, 256, 0, stream>>>(x, W_qkv, qkv, QKV_N, DIMX);

  // 2) pointwise prep
  l2norm_kernel<<<ROWS * 16, 32, 0, stream>>>(qkv);
  silu_kernel<<<(ROWS * 512) / 256, 256, 0, stream>>>(qkv);
  decay_kernel<<<ROWS, 256, 0, stream>>>(x, W_a, A_log, dt_bias, dec);

  // 3) sequential e88 scan: 32 workgroups (one per b,h)
  scan_kernel<<<32, 256, 0, stream>>>(qkv, dec, outs);

  // 4) out = outs @ W_out^T : waves = (8192/16)*(1024/64) = 8192 -> 1024 blocks
  gemm_f32_wmma<<<1024, 256, 0, stream>>>(outs, W_out, out, DIMX, VAL_DIM);
}